// P2RLoss_65257733095794
// MI455X (gfx1250) — compile-verified
//
#include <hip/hip_runtime.h>
#include <math.h>

// ---------------------------------------------------------------------------
// P2RLoss on gfx1250: 1-NN squared distance via V_WMMA_F32_16X16X4_F32
//   d^2 + eps = [u, v, u^2+v^2+eps, 1] . [-2py, -2px, 1, py^2+px^2]^T
// One wave = one 16-pixel tile; loops 128 point tiles (points staged to LDS
// via global_load_async_to_lds_b64). Min-accumulation uses v_min_u32 on raw
// float bits (valid: biased distances are positive), avoiding canonicalize.
// ---------------------------------------------------------------------------

typedef __attribute__((ext_vector_type(2))) float v2f;
typedef __attribute__((ext_vector_type(8))) float v8f;

#define B_             2
#define H_             160
#define W_             160
#define N_             2048
#define HW_            (H_ * W_)
#define NTILES_PTS     (N_ / 16)                       // 128 point tiles
#define TILES_PER_IMG  (HW_ / 16)                      // 1600 pixel tiles/img
#define WAVES_PER_WG   8
#define THREADS_PER_WG (WAVES_PER_WG * 32)
#define WG_PER_IMG     (TILES_PER_IMG / WAVES_PER_WG)  // 200
#define MIN_R2         64.0f                           // MIN_RADIUS^2
#define EPS_BIAS       0.0625f                         // keeps d^2 > 0 in fp32

#if defined(__AMDGCN__) && __has_builtin(__builtin_amdgcn_global_load_async_to_lds_b64)
#define ASYNC_STAGE 1
// Parameter types per hipcc diagnostic: AS1 pointer to int 2-vector (b64).
typedef int v2i_n __attribute__((vector_size(8)));
typedef __attribute__((address_space(1))) v2i_n* async_gptr_t;
typedef __attribute__((address_space(3))) v2i_n* async_lptr_t;
#endif

__global__ void p2r_zero_kernel(float* out) { out[0] = 0.0f; }

__global__ __launch_bounds__(THREADS_PER_WG)
void p2r_wmma_kernel(const float* __restrict__ dens,
                     const float* __restrict__ points,
                     const int* __restrict__ downp,
                     float* __restrict__ out) {
    __shared__ float2 spts[N_];            // 16 KB: all points of this image
    __shared__ float  wsum[WAVES_PER_WG];

    const int tid  = threadIdx.x;
    const int lane = tid & 31;
    const int wave = tid >> 5;
    const int b    = blockIdx.x / WG_PER_IMG;
    const int t    = (blockIdx.x % WG_PER_IMG) * WAVES_PER_WG + wave;

    const float down  = (float)downp[0];
    const float halfc = (down - 1.0f) * 0.5f;

    // ---- Stage this image's points into LDS (async path when available) ----
    const float2* pp = (const float2*)(points + (size_t)b * N_ * 2);
#if ASYNC_STAGE
    for (int i = tid; i < N_; i += THREADS_PER_WG) {
        __builtin_amdgcn_global_load_async_to_lds_b64(
            (async_gptr_t)(pp + i), (async_lptr_t)(spts + i),
            /*offset=*/0, /*cpol=*/0);
    }
#else
    for (int i = tid; i < N_; i += THREADS_PER_WG) spts[i] = pp[i];
#endif

    // Prefetch the density tail (used only after the WMMA loop).
    __builtin_prefetch(dens + (size_t)b * HW_ + t * 16, 0, 1);

    // Pixel tile geometry: 16 consecutive flattened pixels share one row y.
    const int   y   = (t * 16) / W_;
    const int   x0  = (t * 16) % W_;
    const int   m   = lane & 15;                  // pixel/point slot in tile
    const float ccy = (float)y * down;            // tile recentering offsets
    const float ccx = (float)(x0 + 8) * down;

    // A matrix (16x4 f32), row m = [u, v_m, u^2+v_m^2+eps, 1] (centered).
    // Layout (ISA 7.12.2): VGPR0 = {K0 | K2}, VGPR1 = {K1 | K3} per lane half.
    const float u = halfc;                            // same y for whole tile
    const float v = (float)(m - 8) * down + halfc;    // centered x coordinate
    const float s = u * u + v * v + EPS_BIAS;
    v2f amat;
    amat[0] = (lane < 16) ? u : s;
    amat[1] = (lane < 16) ? v : 1.0f;

#if ASYNC_STAGE
#if __has_builtin(__builtin_amdgcn_s_wait_asynccnt)
    __builtin_amdgcn_s_wait_asynccnt(0);
#else
    asm volatile("s_wait_asynccnt 0" ::: "memory");
#endif
#endif
    __syncthreads();

    // Min accumulators as raw float bits (all biased d^2 are positive, so
    // unsigned integer order == float order -> single v_min_u32 per element).
    unsigned int minu[8];
#pragma unroll
    for (int k = 0; k < 8; ++k) minu[k] = 0x7f7fffffu;  // +FLT_MAX bits

    const v8f zacc = {};

    for (int jt = 0; jt < NTILES_PTS; ++jt) {
        float2 p   = spts[jt * 16 + m];
        float  pyc = p.x - ccy;               // points stored (y, x)
        float  pxc = p.y - ccx;
        float  r   = pyc * pyc + pxc * pxc;
        // B matrix (4x16 f32), col j = [-2py, -2px, 1, r]; same VGPR split.
        v2f bmat;
        bmat[0] = (lane < 16) ? (-2.0f * pyc) : 1.0f;
        bmat[1] = (lane < 16) ? (-2.0f * pxc) : r;

        v8f d = __builtin_amdgcn_wmma_f32_16x16x4_f32(
            /*neg_a=*/false, amat, /*neg_b=*/false, bmat,
            /*c_mod=*/(short)0, zacc, /*reuse_a=*/false, /*reuse_b=*/false);
#pragma unroll
        for (int k = 0; k < 8; ++k) {
            unsigned int du = __float_as_uint(d[k]);
            minu[k] = (du < minu[k]) ? du : minu[k];
        }
    }

    // Min over the 16 point columns: xor-shuffle within each 16-lane half.
#pragma unroll
    for (int msk = 8; msk >= 1; msk >>= 1) {
#pragma unroll
        for (int k = 0; k < 8; ++k) {
            unsigned int o =
                (unsigned int)__shfl_xor((int)minu[k], msk, 16);
            minu[k] = (o < minu[k]) ? o : minu[k];
        }
    }

    // Lanes 0 / 16 own per-pixel min d^2 for pixels m=0..7 / 8..15.
    float partial = 0.0f;
    if ((lane & 15) == 0) {
        const float* dbase = dens + (size_t)b * HW_ + t * 16 + ((lane >> 4) << 3);
#pragma unroll
        for (int k = 0; k < 8; ++k) {
            float a  = dbase[k];                                   // logit
            float md = __uint_as_float(minu[k]);                   // d^2 + eps
            float tt = (md < (MIN_R2 + EPS_BIAS)) ? 1.0f : 0.0f;   // target
            float w  = tt + 1.0f;                                  // weight
            float e  = fmaxf(a, 0.0f) - a * tt + log1pf(__expf(-fabsf(a)));
            partial += w * e;
        }
    }
    // Wave sum, then workgroup sum, then one atomic per block.
#pragma unroll
    for (int msk = 16; msk >= 1; msk >>= 1)
        partial += __shfl_xor(partial, msk, 32);
    if (lane == 0) wsum[wave] = partial;
    __syncthreads();
    if (tid == 0) {
        float tot = 0.0f;
#pragma unroll
        for (int wv = 0; wv < WAVES_PER_WG; ++wv) tot += wsum[wv];
        atomicAdd(out, tot * (1.0f / ((float)B_ * (float)HW_)));
    }
}

extern "C" void kernel_launch(void* const* d_in, const int* in_sizes, int n_in,
                              void* d_out, int out_size, void* d_ws, size_t ws_size,
                              hipStream_t stream) {
    const float* dens   = (const float*)d_in[0];
    const float* points = (const float*)d_in[1];
    const int*   down   = (const int*)d_in[2];
    float*       out    = (float*)d_out;

    p2r_zero_kernel<<<1, 1, 0, stream>>>(out);
    p2r_wmma_kernel<<<B_ * WG_PER_IMG, THREADS_PER_WG, 0, stream>>>(
        dens, points, down, out);
}